// Entmax15Loss_15805479649548
// MI455X (gfx1250) — compile-verified
//
#include <hip/hip_runtime.h>
#include <stdint.h>

#define ENTMAX_D_MAX 32000
#define BLK 512
#define NWAVE (BLK / 32)
#define CAP 6144
#define NBISECT 30

// ---------- wave32 primitives ----------
__device__ __forceinline__ float wave_sum(float v) {
#pragma unroll
  for (int off = 16; off > 0; off >>= 1) v += __shfl_xor(v, off, 32);
  return v;
}
__device__ __forceinline__ float wave_max(float v) {
#pragma unroll
  for (int off = 16; off > 0; off >>= 1) v = fmaxf(v, __shfl_xor(v, off, 32));
  return v;
}
__device__ __forceinline__ int wave_iscan_inc(int v) {   // inclusive scan
#pragma unroll
  for (int off = 1; off < 32; off <<= 1) {
    int t = __shfl_up(v, off, 32);
    if ((threadIdx.x & 31) >= off) v += t;
  }
  return v;
}

// 3-barrier generic block reductions (used outside the hot bisection loop)
__device__ __forceinline__ float block_sum(float v, float* scr) {
  const int lane = threadIdx.x & 31, wid = threadIdx.x >> 5;
  v = wave_sum(v);
  if (lane == 0) scr[wid] = v;
  __syncthreads();
  float r = (threadIdx.x < NWAVE) ? scr[threadIdx.x] : 0.0f;
  r = wave_sum(r);
  if (threadIdx.x == 0) scr[0] = r;
  __syncthreads();
  r = scr[0];
  __syncthreads();
  return r;
}
__device__ __forceinline__ float block_max(float v, float* scr) {
  const int lane = threadIdx.x & 31, wid = threadIdx.x >> 5;
  v = wave_max(v);
  if (lane == 0) scr[wid] = v;
  __syncthreads();
  float r = (threadIdx.x < NWAVE) ? scr[threadIdx.x] : -3.402823466e+38f;
  r = wave_max(r);
  if (threadIdx.x == 0) scr[0] = r;
  __syncthreads();
  r = scr[0];
  __syncthreads();
  return r;
}

// ---------- per-row entmax-1.5 loss ----------
__global__ __launch_bounds__(BLK) void entmax15_row_kernel(
    const float* __restrict__ inp, const int* __restrict__ tgt,
    float* __restrict__ row_loss, int d) {
  __shared__ __align__(16) float row[ENTMAX_D_MAX];
  __shared__ float cand[CAP];
  __shared__ float scr[NWAVE];
  __shared__ float rbuf[2][NWAVE];
  __shared__ int   wsum[NWAVE];
  __shared__ int   woff[NWAVE];
  __shared__ int   ktot;

  const int r   = blockIdx.x;
  const int tid = threadIdx.x;
  const int lane = tid & 31, wid = tid >> 5;
  const float* g = inp + (size_t)r * (size_t)d;

  // ---- CDNA5: async bulk copy global -> LDS (ASYNCcnt path, no VGPR staging) ----
  const int d4 = d & ~3;
  for (int i = tid * 4; i < d4; i += BLK * 4) {
    uint32_t laddr = (uint32_t)(uintptr_t)(&row[i]);   // flat low32 == LDS offset
    uint64_t gaddr = (uint64_t)(uintptr_t)(g + i);
    asm volatile("global_load_async_to_lds_b128 %0, %1, off"
                 :: "v"(laddr), "v"(gaddr) : "memory");
  }
  for (int i = d4 + tid; i < d; i += BLK) {
    uint32_t laddr = (uint32_t)(uintptr_t)(&row[i]);
    uint64_t gaddr = (uint64_t)(uintptr_t)(g + i);
    asm volatile("global_load_async_to_lds_b32 %0, %1, off"
                 :: "v"(laddr), "v"(gaddr) : "memory");
  }
  asm volatile("s_wait_asynccnt 0" ::: "memory");
  __syncthreads();

  // ---- pass 1: row max (float4 LDS reads) ----
  const float4* row4 = (const float4*)row;
  const int nchunk = d >> 2;
  float m = -3.402823466e+38f;
  for (int c = tid; c < nchunk; c += BLK) {
    float4 v = row4[c];
    m = fmaxf(m, fmaxf(fmaxf(v.x, v.y), fmaxf(v.z, v.w)));
  }
  for (int i = (nchunk << 2) + tid; i < d; i += BLK) m = fmaxf(m, row[i]);
  m = block_max(m, scr);
  const float mx = 0.5f * m;     // u = 0.5*x - mx, max(u) == 0

  // ---- pass 2: count candidates (u > -1 can ever have clip(u-tau) > 0) ----
  int cnt = 0;
  for (int c = tid; c < nchunk; c += BLK) {
    float4 v = row4[c];
    cnt += (fmaf(0.5f, v.x, -mx) > -1.0f);
    cnt += (fmaf(0.5f, v.y, -mx) > -1.0f);
    cnt += (fmaf(0.5f, v.z, -mx) > -1.0f);
    cnt += (fmaf(0.5f, v.w, -mx) > -1.0f);
  }
  for (int i = (nchunk << 2) + tid; i < d; i += BLK)
    cnt += (fmaf(0.5f, row[i], -mx) > -1.0f);

  // deterministic block exclusive scan of per-thread counts
  const int inc = wave_iscan_inc(cnt);
  if (lane == 31) wsum[wid] = inc;
  __syncthreads();
  if (tid == 0) {
    int acc = 0;
    for (int w = 0; w < NWAVE; ++w) { woff[w] = acc; acc += wsum[w]; }
    ktot = acc;
  }
  __syncthreads();
  const int K = ktot;
  const bool compact = (K <= CAP);   // uniform across block

  // ---- pass 3: ordered compaction (positions are a pure function of data) ----
  if (compact) {
    int off = woff[wid] + (inc - cnt);
    for (int c = tid; c < nchunk; c += BLK) {
      float4 v = row4[c];
      float u;
      u = fmaf(0.5f, v.x, -mx); if (u > -1.0f) cand[off++] = u;
      u = fmaf(0.5f, v.y, -mx); if (u > -1.0f) cand[off++] = u;
      u = fmaf(0.5f, v.z, -mx); if (u > -1.0f) cand[off++] = u;
      u = fmaf(0.5f, v.w, -mx); if (u > -1.0f) cand[off++] = u;
    }
    for (int i = (nchunk << 2) + tid; i < d; i += BLK) {
      float u = fmaf(0.5f, row[i], -mx); if (u > -1.0f) cand[off++] = u;
    }
  }
  __syncthreads();

  // ---- bisection for tau in (-1, 0]: sum clip(u - tau)^2 == 1 ----
  // one barrier per iteration: per-wave partials into a double-buffered array,
  // every wave re-reduces all 16 partials itself (full shfl_xor butterfly).
  float lo = -1.0f, hi = 0.0f;
#pragma unroll 1
  for (int it = 0; it < NBISECT; ++it) {
    const float tau = 0.5f * (lo + hi);
    float s = 0.0f;
    if (compact) {
      for (int i = tid; i < K; i += BLK) {
        float c = fmaxf(cand[i] - tau, 0.0f);
        s = fmaf(c, c, s);
      }
    } else {
      for (int i = tid; i < d; i += BLK) {
        float u = fmaf(0.5f, row[i], -mx);
        float c = fmaxf(u - tau, 0.0f);
        s = fmaf(c, c, s);
      }
    }
    s = wave_sum(s);
    float* sb = rbuf[it & 1];
    if (lane == 0) sb[wid] = s;
    __syncthreads();
    float rr = (lane < NWAVE) ? sb[lane] : 0.0f;
    rr = wave_sum(rr);                 // identical in every wave
    if (rr >= 1.0f) lo = tau; else hi = tau;
  }
  const float tau = 0.5f * (lo + hi);

  // ---- final accumulation: sum p^1.5 = sum c^3, sum p*x ----
  float s15 = 0.0f, spx = 0.0f;
  if (compact) {
    for (int i = tid; i < K; i += BLK) {
      float u = cand[i];
      float c = fmaxf(u - tau, 0.0f);
      float p = c * c;
      float x = 2.0f * (u + mx);
      s15 = fmaf(p, c, s15);
      spx = fmaf(p, x, spx);
    }
  } else {
    for (int i = tid; i < d; i += BLK) {
      float x = row[i];
      float u = fmaf(0.5f, x, -mx);
      float c = fmaxf(u - tau, 0.0f);
      float p = c * c;
      s15 = fmaf(p, c, s15);
      spx = fmaf(p, x, spx);
    }
  }
  s15 = block_sum(s15, scr);
  spx = block_sum(spx, scr);

  if (tid == 0) {
    int t = tgt[r];
    float xt = (t >= 0 && t < d) ? row[t] : 0.0f;      // sum((p - onehot)*x)
    float omega = (1.0f - s15) * (4.0f / 3.0f);        // /0.75
    row_loss[r] = omega + spx - xt;
  }
}

// ---------- deterministic mean over rows ----------
__global__ __launch_bounds__(BLK) void entmax15_reduce_kernel(
    const float* __restrict__ row_loss, float* __restrict__ out, int n) {
  __shared__ float scr[NWAVE];
  float s = 0.0f;
  for (int i = threadIdx.x; i < n; i += BLK) s += row_loss[i];
  s = block_sum(s, scr);
  if (threadIdx.x == 0) out[0] = s / (float)n;
}

extern "C" void kernel_launch(void* const* d_in, const int* in_sizes, int n_in,
                              void* d_out, int out_size, void* d_ws, size_t ws_size,
                              hipStream_t stream) {
  const float* inp = (const float*)d_in[0];
  const int*   tgt = (const int*)d_in[1];
  float* out      = (float*)d_out;
  float* row_loss = (float*)d_ws;

  const int n = in_sizes[1];                 // 4096 rows
  const int d = in_sizes[0] / n;             // 32000 classes (<= ENTMAX_D_MAX)

  entmax15_row_kernel<<<n, BLK, 0, stream>>>(inp, tgt, row_loss, d);
  entmax15_reduce_kernel<<<1, BLK, 0, stream>>>(row_loss, out, n);
}